// NetworkClassPyTorch_48215302865321
// MI455X (gfx1250) — compile-verified
//
#include <hip/hip_runtime.h>
#include <math.h>

typedef __attribute__((ext_vector_type(2))) float v2f;
typedef __attribute__((ext_vector_type(8))) float v8f;
typedef __attribute__((ext_vector_type(4))) unsigned int v4u;
typedef __attribute__((ext_vector_type(4))) int v4i;
typedef __attribute__((ext_vector_type(8))) int v8i;

#define BATCH 32
#define N_IN  64
#define NN    1024
#define ALPHA 0.995f   // 1 - 0.05/10
#define BETA  0.975f   // 1 - 0.05/2
#define V_TH  2.0f

// ---------------------------------------------------------------------------
// Kernel 1: elementwise prep (streams the 8MB x tensor once, float4 loads).
//   zon[b,k] = BETA*z_out + z                      -> z_out_new output slot
//   zsh[b,k] = zon[b,k+1] (zsh[b,1023] = 0)        -> workspace
//   acc[b,k] = ALPHA*v - V_TH*z + sum_{j<64} x[b,j,k]*w[j,k]  -> workspace
// ---------------------------------------------------------------------------
__global__ void __launch_bounds__(256)
prep_kernel(const float* __restrict__ x,
            const float* __restrict__ v,
            const float* __restrict__ z,
            const float* __restrict__ z_out,
            const float* __restrict__ w,
            float* __restrict__ zon_out,
            float* __restrict__ zsh_ws,
            float* __restrict__ acc_ws) {
    int t  = blockIdx.x * blockDim.x + threadIdx.x;   // 0 .. 8191
    int b  = t >> 8;
    int kq = (t & 255) << 2;                          // aligned group of 4 cols
    int base = b * NN + kq;

    float4 zv = *(const float4*)(z + base);
    float4 zo = *(const float4*)(z_out + base);
    float4 vv = *(const float4*)(v + base);

    float4 zon;
    zon.x = BETA * zo.x + zv.x;
    zon.y = BETA * zo.y + zv.y;
    zon.z = BETA * zo.z + zv.z;
    zon.w = BETA * zo.w + zv.w;
    *(float4*)(zon_out + base) = zon;

    float4 zsh;
    zsh.x = zon.y;
    zsh.y = zon.z;
    zsh.z = zon.w;
    zsh.w = (kq == NN - 4) ? 0.0f : (BETA * z_out[base + 4] + z[base + 4]);
    *(float4*)(zsh_ws + base) = zsh;

    float4 acc;
    acc.x = ALPHA * vv.x - V_TH * zv.x;
    acc.y = ALPHA * vv.y - V_TH * zv.y;
    acc.z = ALPHA * vv.z - V_TH * zv.z;
    acc.w = ALPHA * vv.w - V_TH * zv.w;

    const float* xp = x + (size_t)b * N_IN * NN + kq;
    const float* wp = w + kq;
#pragma unroll 4
    for (int j = 0; j < N_IN; ++j) {
        float4 xv = *(const float4*)(xp + (size_t)j * NN);
        float4 wv = *(const float4*)(wp + (size_t)j * NN);
        acc.x += xv.x * wv.x;
        acc.y += xv.y * wv.y;
        acc.z += xv.z * wv.z;
        acc.w += xv.w * wv.w;
    }
    *(float4*)(acc_ws + base) = acc;
}

// ---------------------------------------------------------------------------
// Kernel 2: TDM-staged recurrent GEMM via V_WMMA_F32_16X16X4_F32.
//   rec[b,k] = sum_{m<k} zon[b,m]*wrec[m,k] + sum_{m>=k} zsh[b,m]*wrec[m,k]
// Per block: one TENSOR_LOAD_TO_LDS DMAs the 1024x16 column slab of wrec
// (64KB) into LDS; 8 waves each own a 128-wide K slice of the K loop and
// both batch-tile accumulators. Three scalar-uniform loop regions (below /
// overlap / above the diagonal block) keep the inner loops branch-free;
// B fragments come from LDS (conflict-free), A fragments are b64 global
// loads of zon/zsh (L2-resident, 128KB each). Partials reduced via LDS.
// ---------------------------------------------------------------------------
__global__ void __launch_bounds__(256)
wmma_gemm_kernel(const float* __restrict__ w,
                 const float* __restrict__ zon,
                 const float* __restrict__ zsh,
                 const float* __restrict__ acc,
                 float* __restrict__ v_out,
                 float* __restrict__ z_new_out,
                 float* __restrict__ r_out) {
    __shared__ float lds_b[NN * 16];   // 64KB: wrec[m][k0..k0+15], row-major
    __shared__ float red[8 * 512];     // 16KB: per-wave partial tiles

    const int k0   = blockIdx.x * 16;
    const int tid  = threadIdx.x;
    const int wv   = tid >> 5;         // wave id 0..7
    const int lane = tid & 31;
    const int half = lane >> 4;
    const int l    = lane & 15;
    const int n    = k0 + l;           // this lane's output column

    const float* wrec = w + (size_t)N_IN * NN;      // wrec[m*NN+k] = w[64+m,k]

    // ---- TDM: DMA the 1024x16 B slab into LDS (wave 0 issues) ----
    if (wv == 0) {
        unsigned long long gaddr = (unsigned long long)(const void*)(wrec + k0);
        unsigned int lds_off = (unsigned int)(unsigned long long)(void*)&lds_b[0];

        v4u g0;
        g0.x = 1u;                                   // count=1, no gather
        g0.y = lds_off;                              // lds_addr (bytes)
        g0.z = (unsigned int)(gaddr & 0xFFFFFFFFull);
        g0.w = (unsigned int)((gaddr >> 32) & 0x01FFFFFFull) | (2u << 30); // type=2

        v8i g1;
        g1[0] = (int)(2u << 16);                     // data_size=2 (4B), mask=0
        g1[1] = (int)(16u << 16);                    // tensor_dim0[15:0]=16
        g1[2] = (int)(1024u << 16);                  // tensor_dim1[15:0]=1024
        g1[3] = (int)(16u << 16);                    // tile_dim0=16
        g1[4] = (int)(1024u);                        // tile_dim1=1024, tile_dim2=0
        g1[5] = (int)(1024u);                        // tensor_dim0_stride=1024
        g1[6] = 0;                                   // stride0 hi, stride1 lo
        g1[7] = 0;                                   // stride1 hi

        v4i g2 = {0, 0, 0, 0};
        v4i g3 = {0, 0, 0, 0};
        v8i g4 = {0, 0, 0, 0, 0, 0, 0, 0};

        __builtin_amdgcn_tensor_load_to_lds(g0, g1, g2, g3, g4, 0);
        __builtin_amdgcn_s_wait_tensorcnt(0);
    }
    __syncthreads();

    const float* za0 = zon + (size_t)l * NN;        // batch rows 0..15
    const float* za1 = zon + (size_t)(l + 16) * NN;
    const float* zs0 = zsh + (size_t)l * NN;
    const float* zs1 = zsh + (size_t)(l + 16) * NN;

    // scalarize wave-uniform K-slice bounds -> SALU loops, EXEC untouched
    const int Ks = __builtin_amdgcn_readfirstlane(wv * (NN / 8));
    const int Ke = Ks + (NN / 8);
    const int e1 = (Ke < k0) ? Ke : k0;              // zon region:  K < k0
    const int s2 = (Ks > k0) ? Ks : k0;              // overlap:     k0 <= K < k0+16
    const int e2 = (Ke < k0 + 16) ? Ke : (k0 + 16);
    const int s3 = (Ks > k0 + 16) ? Ks : (k0 + 16);  // zsh region:  K >= k0+16

    v8f c0 = {};   // batch 0..15
    v8f c1 = {};   // batch 16..31

    // ---- region 1: all m < k -> zon with unmasked B (from LDS) ----
    for (int K = Ks; K < e1; K += 4) {
        const int kk0 = K + 2 * half;
        v2f a0 = *(const v2f*)(za0 + kk0);
        v2f a1 = *(const v2f*)(za1 + kk0);
        v2f bf;
        bf.x = lds_b[kk0 * 16 + l];
        bf.y = lds_b[(kk0 + 1) * 16 + l];
        c0 = __builtin_amdgcn_wmma_f32_16x16x4_f32(false, a0, false, bf,
                                                   (short)0, c0, false, false);
        c1 = __builtin_amdgcn_wmma_f32_16x16x4_f32(false, a1, false, bf,
                                                   (short)0, c1, false, false);
    }

    // ---- region 2: overlap -> masked complementary B fragments ----
    for (int K = s2; K < e2; K += 4) {
        const int kk0 = K + 2 * half;
        const int kk1 = kk0 + 1;
        float w0 = lds_b[kk0 * 16 + l];
        float w1 = lds_b[kk1 * 16 + l];
        v2f b1, b2;
        b1.x = (kk0 < n) ? w0 : 0.0f;      // m <  k part (zon)
        b1.y = (kk1 < n) ? w1 : 0.0f;
        b2.x = w0 - b1.x;                  // m >= k part (zsh), exact
        b2.y = w1 - b1.y;
        v2f az0 = *(const v2f*)(za0 + kk0);
        v2f az1 = *(const v2f*)(za1 + kk0);
        v2f as0 = *(const v2f*)(zs0 + kk0);
        v2f as1 = *(const v2f*)(zs1 + kk0);
        c0 = __builtin_amdgcn_wmma_f32_16x16x4_f32(false, az0, false, b1,
                                                   (short)0, c0, false, false);
        c0 = __builtin_amdgcn_wmma_f32_16x16x4_f32(false, as0, false, b2,
                                                   (short)0, c0, false, false);
        c1 = __builtin_amdgcn_wmma_f32_16x16x4_f32(false, az1, false, b1,
                                                   (short)0, c1, false, false);
        c1 = __builtin_amdgcn_wmma_f32_16x16x4_f32(false, as1, false, b2,
                                                   (short)0, c1, false, false);
    }

    // ---- region 3: all m >= k -> zsh with unmasked B (from LDS) ----
    for (int K = s3; K < Ke; K += 4) {
        const int kk0 = K + 2 * half;
        v2f a0 = *(const v2f*)(zs0 + kk0);
        v2f a1 = *(const v2f*)(zs1 + kk0);
        v2f bf;
        bf.x = lds_b[kk0 * 16 + l];
        bf.y = lds_b[(kk0 + 1) * 16 + l];
        c0 = __builtin_amdgcn_wmma_f32_16x16x4_f32(false, a0, false, bf,
                                                   (short)0, c0, false, false);
        c1 = __builtin_amdgcn_wmma_f32_16x16x4_f32(false, a1, false, bf,
                                                   (short)0, c1, false, false);
    }

    // ---- cross-wave reduction through LDS ----
#pragma unroll
    for (int r = 0; r < 8; ++r) {
        const int e = (8 * half + r) * 16 + l;      // element id within tile
        red[wv * 512 + e]       = c0[r];
        red[wv * 512 + 256 + e] = c1[r];
    }
    __syncthreads();

    // ---- finalize: 256 threads, one (M, Ncol) element of both batch tiles --
    const int M  = tid >> 4;
    const int Nc = tid & 15;
    const int k  = k0 + Nc;

    float s0 = 0.0f, s1 = 0.0f;
#pragma unroll
    for (int wi = 0; wi < 8; ++wi) {
        s0 += red[wi * 512 + tid];
        s1 += red[wi * 512 + 256 + tid];
    }

    const float cw = 10.0f / 31.0f;
    float dx = (float)(k & 31) * cw - 5.0f;
    float dy = (float)(k >> 5) * cw - 5.0f;
    float rbase = 1.0f + expf(-0.5f * (dx * dx + dy * dy));

    int idx0 = M * NN + k;                 // batch rows 0..15
    float vn0 = acc[idx0] + s0;
    float zn0 = (vn0 - V_TH > 0.0f) ? 1.0f : 0.0f;
    v_out[idx0]     = vn0;
    z_new_out[idx0] = zn0;
    r_out[idx0]     = rbase * ((zn0 > 0.0f) ? 1.5f : 1.0f);

    int idx1 = (16 + M) * NN + k;          // batch rows 16..31
    float vn1 = acc[idx1] + s1;
    float zn1 = (vn1 - V_TH > 0.0f) ? 1.0f : 0.0f;
    v_out[idx1]     = vn1;
    z_new_out[idx1] = zn1;
    r_out[idx1]     = rbase * ((zn1 > 0.0f) ? 1.5f : 1.0f);
}

// ---------------------------------------------------------------------------
// Launch. Inputs (setup_inputs order): x, v, z, z_out, w.
// d_out = [v_new | z_new | z_out_new | R], each 32*1024 f32.
// Workspace: acc (128KB) + zsh (128KB).
// ---------------------------------------------------------------------------
extern "C" void kernel_launch(void* const* d_in, const int* in_sizes, int n_in,
                              void* d_out, int out_size, void* d_ws, size_t ws_size,
                              hipStream_t stream) {
    const float* x     = (const float*)d_in[0];
    const float* v     = (const float*)d_in[1];
    const float* z     = (const float*)d_in[2];
    const float* z_out = (const float*)d_in[3];
    const float* w     = (const float*)d_in[4];

    float* out       = (float*)d_out;
    float* v_new     = out;                       // [32,1024]
    float* z_new     = out + BATCH * NN;          // [32,1024]
    float* z_out_new = out + 2 * BATCH * NN;      // [32,1024]
    float* R         = out + 3 * BATCH * NN;      // [32,1024]

    float* acc_ws = (float*)d_ws;                 // 32*1024 floats
    float* zsh_ws = (float*)d_ws + BATCH * NN;    // 32*1024 floats

    prep_kernel<<<(BATCH * NN / 4) / 256, 256, 0, stream>>>(
        x, v, z, z_out, w, z_out_new, zsh_ws, acc_ws);

    wmma_gemm_kernel<<<NN / 16, 256, 0, stream>>>(
        w, z_out_new, zsh_ws, acc_ws, v_new, z_new, R);
}